// PVRPInitEmbedding_23725399343391
// MI455X (gfx1250) — compile-verified
//
#include <hip/hip_runtime.h>
#include <hip/hip_bf16.h>
#include <math.h>

typedef __attribute__((ext_vector_type(2))) float v2f;
typedef __attribute__((ext_vector_type(4))) float v4f;
typedef __attribute__((ext_vector_type(8))) float v8f;

#define BB 32
#define MM 10
#define NN 1000
#define DD 128
#define NODES 1010            // M + N
#define ROWS (BB * NODES)     // 32320 rows of nodes_embedding / NEproj

// ---------------------------------------------------------------------------
// A0: peproj[a][d] = (pe @ W_posproj.T)[a][d]   (batch-invariant, 10x128)
// ---------------------------------------------------------------------------
__global__ void k_peproj(const float* __restrict__ Wpp, float* __restrict__ peproj) {
    int a = blockIdx.x, d = threadIdx.x;
    const float c = -9.210340371976184f / 128.0f;   // -ln(10000)/128
    float s = 0.0f;
    for (int i = 0; i < 64; ++i) {
        float ang = (float)a * expf((float)(2 * i) * c);
        s += sinf(ang) * Wpp[d * 128 + 2 * i] + cosf(ang) * Wpp[d * 128 + 2 * i + 1];
    }
    peproj[a * 128 + d] = s;
}

// ---------------------------------------------------------------------------
// A3: v_pref[j] = (Wp @ W_pref[:,0])[j],  Wp = W_final[:,128:256]
// ---------------------------------------------------------------------------
__global__ void k_vpref(const float* __restrict__ Wf, const float* __restrict__ Wpref,
                        float* __restrict__ vpref) {
    int j = threadIdx.x;
    float s = 0.0f;
    for (int k = 0; k < 128; ++k) s += Wf[j * 384 + 128 + k] * Wpref[k];
    vpref[j] = s;
}

// ---------------------------------------------------------------------------
// A1: per (b,a): depot_embedding -> ws, agents_embedding -> nodes output
// ---------------------------------------------------------------------------
__global__ void k_agents(const float* __restrict__ locs, const float* __restrict__ cap,
                         const float* __restrict__ spd, const float* __restrict__ Wdep,
                         const float* __restrict__ alpha, const float* __restrict__ Wag,
                         const float* __restrict__ peproj,
                         float* __restrict__ depot_emb, float* __restrict__ nodes) {
    int ba = blockIdx.x, b = ba / MM, a = ba % MM, d = threadIdx.x;
    float x = locs[((size_t)b * NODES + a) * 2 + 0];
    float y = locs[((size_t)b * NODES + a) * 2 + 1];
    depot_emb[(size_t)ba * DD + d] =
        x * Wdep[d * 2] + y * Wdep[d * 2 + 1] + alpha[0] * peproj[a * 128 + d];
    float cp = cap[ba] * (1.0f / 40.0f);
    float sp = spd[ba];
    nodes[((size_t)b * NODES + a) * DD + d] =
        x * Wag[d * 4] + y * Wag[d * 4 + 1] + cp * Wag[d * 4 + 2] + sp * Wag[d * 4 + 3];
}

// ---------------------------------------------------------------------------
// B: clients_embedding -> nodes output rows M..M+N-1
// ---------------------------------------------------------------------------
__global__ void k_clients(const float* __restrict__ locs, const float* __restrict__ demand,
                          const float* __restrict__ Wc, float* __restrict__ nodes) {
    int bn = blockIdx.x, b = bn / NN, n = bn % NN, node = MM + n, d = threadIdx.x;
    float x  = locs[((size_t)b * NODES + node) * 2 + 0];
    float y  = locs[((size_t)b * NODES + node) * 2 + 1];
    float dx = x - locs[((size_t)b * NODES) * 2 + 0];
    float dy = y - locs[((size_t)b * NODES) * 2 + 1];
    float dist = sqrtf(dx * dx + dy * dy);
    float ang  = atan2f(dy, dx);
    float dm   = demand[(size_t)b * NODES + node] * (1.0f / 40.0f);
    nodes[((size_t)b * NODES + node) * DD + d] =
        x * Wc[d * 5] + y * Wc[d * 5 + 1] + dm * Wc[d * 5 + 2] +
        dist * Wc[d * 5 + 3] + ang * Wc[d * 5 + 4];
}

// ---------------------------------------------------------------------------
// A2: DAd[b,a,:] = Wd @ (W_depot_agents @ [depot_emb; agents_emb])
// ---------------------------------------------------------------------------
__global__ void k_dad(const float* __restrict__ depot_emb, const float* __restrict__ nodes,
                      const float* __restrict__ Wda, const float* __restrict__ Wf,
                      float* __restrict__ DAd) {
    __shared__ float xin[256];
    __shared__ float da[128];
    int ba = blockIdx.x, b = ba / MM, a = ba % MM, t = threadIdx.x;
    xin[t]       = depot_emb[(size_t)ba * DD + t];
    xin[128 + t] = nodes[((size_t)b * NODES + a) * DD + t];
    __syncthreads();
    float s = 0.0f;
    for (int k = 0; k < 256; ++k) s += Wda[t * 256 + k] * xin[k];
    da[t] = s;
    __syncthreads();
    float s2 = 0.0f;
    for (int k = 0; k < 128; ++k) s2 += Wf[t * 384 + 256 + k] * da[k];
    DAd[(size_t)ba * DD + t] = s2;
}

// ---------------------------------------------------------------------------
// C: NEproj = nodes_embedding @ Wn.T via V_WMMA_F32_16X16X4_F32.
// Each wave: 16 rows x 128 cols, 32 K-steps x 8 col-tiles = 256 WMMA.
//
// Latency strategy (fix for serialized load->wait->wmma):
//   * Entire A strip for this lane preloaded into 32 v2f regs (one wait).
//   * B fragments double-buffered in registers: the 8 loads for k-step ks+1
//     are issued while the 8 WMMAs for ks execute, so the steady state keeps
//     8 loads in flight and WMMAs run back-to-back.
// ---------------------------------------------------------------------------
__global__ void k_neproj(const float* __restrict__ NE, const float* __restrict__ Wf,
                         float* __restrict__ nep, int nepSB) {
    int lane = threadIdx.x & 31;
    int half = lane >> 4;
    int l15  = lane & 15;
    int w = (blockIdx.x * blockDim.x + threadIdx.x) >> 5;   // uniform per wave
    if (w >= ROWS / 16) return;
    int r0 = w * 16;

    const float* Arow  = NE + (size_t)(r0 + l15) * DD;   // this lane's A row
    const float* Bbase = Wf + (size_t)l15 * 384 + 2 * half; // Wn row j = out col

    // Preload all A fragments: lane's K elements are pairs at 4*ks + 2*half.
    v2f a[32];
#pragma unroll
    for (int ks = 0; ks < 32; ++ks)
        a[ks] = *(const v2f*)(Arow + 4 * ks + 2 * half);

    v8f acc[8] = {};
    v2f bc[8], bn[8];
#pragma unroll
    for (int ct = 0; ct < 8; ++ct)
        bc[ct] = *(const v2f*)(Bbase + (size_t)ct * 16 * 384);

#pragma unroll
    for (int ks = 0; ks < 32; ++ks) {
        if (ks + 1 < 32) {
#pragma unroll
            for (int ct = 0; ct < 8; ++ct)
                bn[ct] = *(const v2f*)(Bbase + (size_t)ct * 16 * 384 + 4 * (ks + 1));
        }
#pragma unroll
        for (int ct = 0; ct < 8; ++ct)
            acc[ct] = __builtin_amdgcn_wmma_f32_16x16x4_f32(
                false, a[ks], false, bc[ct], (short)0, acc[ct], false, false);
#pragma unroll
        for (int ct = 0; ct < 8; ++ct) bc[ct] = bn[ct];
    }

    // D tile: VGPR v -> row r0 + v + half*8, col = ct*16 + l15
#pragma unroll
    for (int v = 0; v < 8; ++v) {
        int row = r0 + v + half * 8;
        int b = row / NODES;
        int n = row - b * NODES;
        float* out = nep + (size_t)b * nepSB + (size_t)n * DD + l15;
#pragma unroll
        for (int ct = 0; ct < 8; ++ct) out[ct * 16] = acc[ct][v];
    }
}

// ---------------------------------------------------------------------------
// D: combined[b,a,n,:] = NEproj[b,n,:] + pref[b,a,n]*v_pref + DAd[b,a,:]
// One wave per (b,n) row; float4 traffic; NT stores for the 165MB output.
// nep may alias the a=0 slice of comb (fallback) -> no __restrict__, read first.
// ---------------------------------------------------------------------------
__global__ void k_combined(const float* nep, int nepSB,
                           const float* __restrict__ DAd, const float* __restrict__ vpref,
                           const float* __restrict__ pref, float* comb) {
    int tx  = threadIdx.x;                                // 0..31 -> d4 group
    int row = blockIdx.x * blockDim.y + threadIdx.y;
    if (row >= ROWS) return;
    int b = row / NODES;
    int n = row - b * NODES;
    v4f ne = *(const v4f*)(nep + (size_t)b * nepSB + (size_t)n * DD + tx * 4);
    v4f vp = *(const v4f*)(vpref + tx * 4);
#pragma unroll
    for (int a = 0; a < MM; ++a) {
        float p = pref[((size_t)b * MM + a) * NODES + n];
        v4f da = *(const v4f*)(DAd + ((size_t)b * MM + a) * DD + tx * 4);
        v4f o = ne + p * vp + da;
        __builtin_nontemporal_store(
            o, (v4f*)(comb + (((size_t)b * MM + a) * NODES + n) * (size_t)DD + tx * 4));
    }
}

// ---------------------------------------------------------------------------
extern "C" void kernel_launch(void* const* d_in, const int* in_sizes, int n_in,
                              void* d_out, int out_size, void* d_ws, size_t ws_size,
                              hipStream_t stream) {
    const float* locs     = (const float*)d_in[0];
    const float* capacity = (const float*)d_in[1];
    const float* speed    = (const float*)d_in[2];
    const float* demand   = (const float*)d_in[3];
    const float* pref     = (const float*)d_in[4];
    // d_in[5] action_mask: shape only, unused
    const float* Wdep  = (const float*)d_in[6];
    const float* Wpp   = (const float*)d_in[7];
    const float* alpha = (const float*)d_in[8];
    const float* Wag   = (const float*)d_in[9];
    const float* Wda   = (const float*)d_in[10];
    const float* Wc    = (const float*)d_in[11];
    const float* Wpref = (const float*)d_in[12];
    const float* Wf    = (const float*)d_in[13];

    float* nodes = (float*)d_out;                   // output 1: (32320,128)
    float* comb  = nodes + (size_t)ROWS * DD;       // output 2: (B,M,NODES,D)

    // small workspace (~330 KB)
    float* depot_emb = (float*)d_ws;                // 320*128
    float* DAd       = depot_emb + BB * MM * DD;    // 320*128
    float* peproj    = DAd + BB * MM * DD;          // 10*128
    float* vpref     = peproj + MM * DD;            // 128
    float* wsend     = vpref + DD;
    size_t smallBytes = (size_t)(wsend - (float*)d_ws) * sizeof(float);
    size_t nepBytes   = (size_t)ROWS * DD * sizeof(float);

    float* nep;
    int nepSB;
    if (ws_size >= smallBytes + nepBytes) {
        nep   = wsend;                 // NEproj in workspace
        nepSB = NODES * DD;
    } else {
        nep   = comb;                  // stage NEproj in a=0 slice of combined
        nepSB = MM * NODES * DD;       // (k_combined reads before it writes)
    }

    k_peproj <<<MM, DD, 0, stream>>>(Wpp, peproj);
    k_vpref  <<<1, DD, 0, stream>>>(Wf, Wpref, vpref);
    k_agents <<<BB * MM, DD, 0, stream>>>(locs, capacity, speed, Wdep, alpha, Wag,
                                          peproj, depot_emb, nodes);
    k_clients<<<BB * NN, DD, 0, stream>>>(locs, demand, Wc, nodes);
    k_dad    <<<BB * MM, DD, 0, stream>>>(depot_emb, nodes, Wda, Wf, DAd);

    // 2020 waves, 4 waves (128 threads) per block -> 505 blocks
    k_neproj <<<(ROWS / 16) / 4, 128, 0, stream>>>(nodes, Wf, nep, nepSB);

    dim3 blk(32, 8);                                // one wave per (b,n) row
    k_combined<<<ROWS / 8, blk, 0, stream>>>(nep, nepSB, DAd, vpref, pref, comb);
}